// SimplifiedOptimizedFractionalActivation_58677843198091
// MI455X (gfx1250) — compile-verified
//
#include <hip/hip_runtime.h>
#include <stdint.h>

// ---------------------------------------------------------------------------
// FractionalActivation for MI455X (gfx1250, wave32).
//   pass 0: init min/max accumulators in d_ws (must re-init every launch)
//   pass 1: global min/max reduction (b128 loads, wave32 shuffle + LDS + atomics)
//   pass 2: pointwise map, tiles walked in REVERSE of pass-1 order for L2 reuse,
//           input staged via async global->LDS double buffering (ASYNCcnt),
//           output written with non-temporal stores.
// No WMMA: the op has no matmul structure; it is HBM-bandwidth bound (~34us
// theoretical at 23.3 TB/s for 805MB of traffic). CDNA5-specific paths used:
// global_load_async_to_lds_b128 / s_wait_asynccnt / s_wait_dscnt / NT stores.
// ---------------------------------------------------------------------------

typedef float v4f __attribute__((ext_vector_type(4)));
typedef int   v4i __attribute__((ext_vector_type(4)));

#define MIN_STEP 1e-06f
#define MAX_STEP 1e-03f

#ifndef __has_builtin
#define __has_builtin(x) 0
#endif

// Async-to-LDS path only exists in the gfx1250 device pass.
#if defined(__gfx1250__) && __has_builtin(__builtin_amdgcn_global_load_async_to_lds_b128)
#define USE_ASYNC 1
#else
#define USE_ASYNC 0
#endif

#if USE_ASYNC
#if __has_builtin(__builtin_amdgcn_s_wait_asynccnt)
#define WAIT_ASYNC(n)                                \
  do {                                               \
    __builtin_amdgcn_s_wait_asynccnt(n);             \
    asm volatile("" ::: "memory");                   \
  } while (0)
#else
#define WAIT_ASYNC(n) asm volatile("s_wait_asynccnt " #n ::: "memory")
#endif
// Builtin signature (from clang diagnostic): (int4 AS1*, int4 AS3*, imm, imm).
typedef v4i __attribute__((address_space(1))) g4i;  // global int4
typedef v4i __attribute__((address_space(3))) s4i;  // LDS int4
#endif

// Order-preserving float <-> uint32 key (for integer atomic min/max).
__device__ __forceinline__ unsigned f2key(float f) {
  unsigned b = __float_as_uint(f);
  return (b & 0x80000000u) ? ~b : (b | 0x80000000u);
}
__device__ __forceinline__ float key2f(unsigned k) {
  unsigned b = (k & 0x80000000u) ? (k ^ 0x80000000u) : ~k;
  return __uint_as_float(b);
}

// ---------------------------------------------------------------------------
// Pass 0: reset accumulators (d_ws is poisoned once and never re-poisoned).
// ---------------------------------------------------------------------------
__global__ void fa_init_minmax(unsigned* mm) {
  if (threadIdx.x == 0) {
    mm[0] = 0xFFFFFFFFu;  // running MIN key (start at +max key)
    mm[1] = 0x00000000u;  // running MAX key (start at min key)
  }
}

// ---------------------------------------------------------------------------
// Pass 1: global min/max.
// ---------------------------------------------------------------------------
__global__ void __launch_bounds__(256)
fa_minmax_reduce(const float* __restrict__ in, unsigned* __restrict__ mm, long n) {
  const long n4 = n >> 2;
  const v4f* __restrict__ in4 = (const v4f*)in;

  float vmin = __builtin_inff();
  float vmax = -__builtin_inff();

  const long stride = (long)gridDim.x * blockDim.x;
  for (long i = (long)blockIdx.x * blockDim.x + threadIdx.x; i < n4; i += stride) {
    v4f v = in4[i];  // global_load_b128
    vmin = fminf(vmin, fminf(fminf(v[0], v[1]), fminf(v[2], v[3])));
    vmax = fmaxf(vmax, fmaxf(fmaxf(v[0], v[1]), fmaxf(v[2], v[3])));
  }
  // scalar tail (n not divisible by 4), block 0 only
  if (blockIdx.x == 0) {
    for (long i = (n4 << 2) + threadIdx.x; i < n; i += blockDim.x) {
      float x = in[i];
      vmin = fminf(vmin, x);
      vmax = fmaxf(vmax, x);
    }
  }

  // wave32 butterfly reduction
#pragma unroll
  for (int off = 16; off > 0; off >>= 1) {
    vmin = fminf(vmin, __shfl_xor(vmin, off, 32));
    vmax = fmaxf(vmax, __shfl_xor(vmax, off, 32));
  }

  __shared__ float smin[8];
  __shared__ float smax[8];
  const int wave = threadIdx.x >> 5;
  const int lane = threadIdx.x & 31;
  if (lane == 0) {
    smin[wave] = vmin;
    smax[wave] = vmax;
  }
  __syncthreads();
  if (threadIdx.x == 0) {
    float bmin = smin[0], bmax = smax[0];
#pragma unroll
    for (int w = 1; w < 8; ++w) {
      bmin = fminf(bmin, smin[w]);
      bmax = fmaxf(bmax, smax[w]);
    }
    atomicMin(&mm[0], f2key(bmin));  // global_atomic_min_u32
    atomicMax(&mm[1], f2key(bmax));  // global_atomic_max_u32
  }
}

// ---------------------------------------------------------------------------
// Pointwise math. 4 transcendentals/element instead of 6:
//   h   = MIN_STEP + (x - xmin) * scale
//   e   = exp(-x),  t = 1 + e,  sigmoid = rcp(t)
//   p   = (1 - e^-h)/h = 1 - h/2 + h^2/6      (truncation error <5e-11 for h<=1e-3)
//   fd  = (sig(x+h)-sig(x))/h = e*p / ( t * (t - e*h*p) )   -- no cancellation
//   out = sigmoid + sqrt(fd);  NaN(x) passes through.
// ---------------------------------------------------------------------------
__device__ __forceinline__ float frac_act(float x, float xmin, float scale) {
  float h = fmaf(x - xmin, scale, MIN_STEP);
  float e = __expf(-x);                       // v_exp_f32
  float t = 1.0f + e;
  float s = __builtin_amdgcn_rcpf(t);         // v_rcp_f32 : sigmoid(x)
  float p = 1.0f - h * (0.5f - h * (1.0f / 6.0f));
  float d = t * fmaf(-e, h * p, t);           // t * (t - e*h*p)
  float ratio = (e * p) * __builtin_amdgcn_rcpf(d);
  float r = s + __builtin_amdgcn_sqrtf(ratio);  // v_sqrt_f32
  return (x != x) ? x : r;
}

// ---------------------------------------------------------------------------
// Pass 2: map. Tiles (256 x float4 = 4KB) walked in reverse so the data pass 1
// touched last (still hot in the 192MB L2) is consumed first.
// ---------------------------------------------------------------------------
__global__ void __launch_bounds__(256)
fa_map(const float* __restrict__ in, float* __restrict__ out,
       const unsigned* __restrict__ mm, long n) {
  const float xmin = key2f(mm[0]);
  const float xmax = key2f(mm[1]);
  const float scale = (MAX_STEP - MIN_STEP) / (xmax - xmin);

  const long n4 = n >> 2;
  const long tiles = n4 >> 8;  // 256 float4 per tile
  const v4f* __restrict__ in4 = (const v4f*)in;
  v4f* __restrict__ out4 = (v4f*)out;
  const int tid = threadIdx.x;

#if USE_ASYNC
  // Double-buffered async global->LDS pipeline (ASYNCcnt-tracked DMA path).
  __shared__ v4f stage[2][256];  // 8KB

  long i = blockIdx.x;
  if (i < tiles) {
    const long t0 = tiles - 1 - i;
    __builtin_amdgcn_global_load_async_to_lds_b128(
        (g4i*)(in4 + (t0 << 8) + tid), (s4i*)&stage[0][tid], 0, 0);
  }
  int buf = 0;
  for (; i < tiles; i += gridDim.x) {
    const long inext = i + (long)gridDim.x;
    if (inext < tiles) {
      const long tn = tiles - 1 - inext;
      // make sure our previous ds_load of this buffer fully retired before
      // the async engine is allowed to overwrite it
      asm volatile("s_wait_dscnt 0x0" ::: "memory");
      __builtin_amdgcn_global_load_async_to_lds_b128(
          (g4i*)(in4 + (tn << 8) + tid), (s4i*)&stage[buf ^ 1][tid], 0, 0);
      WAIT_ASYNC(1);  // oldest (current buffer) complete; loads finish in order
    } else {
      WAIT_ASYNC(0);
    }
    v4f v = stage[buf][tid];  // ds_load_b128
    v4f r;
    r[0] = frac_act(v[0], xmin, scale);
    r[1] = frac_act(v[1], xmin, scale);
    r[2] = frac_act(v[2], xmin, scale);
    r[3] = frac_act(v[3], xmin, scale);
    const long t = tiles - 1 - i;
    __builtin_nontemporal_store(r, &out4[(t << 8) + tid]);  // TH=NT b128 store
    buf ^= 1;
  }
#else
  // Fallback: plain reversed grid-stride over float4s.
  const long total = tiles << 8;
  const long stride = (long)gridDim.x * blockDim.x;
  for (long i = (long)blockIdx.x * blockDim.x + tid; i < total; i += stride) {
    const long j = total - 1 - i;
    v4f v = in4[j];
    v4f r;
    r[0] = frac_act(v[0], xmin, scale);
    r[1] = frac_act(v[1], xmin, scale);
    r[2] = frac_act(v[2], xmin, scale);
    r[3] = frac_act(v[3], xmin, scale);
    __builtin_nontemporal_store(r, &out4[j]);
  }
#endif

  // scalar tail: everything past the tiled region
  if (blockIdx.x == 0) {
    for (long idx = (tiles << 10) + tid; idx < n; idx += blockDim.x) {
      const float x = in[idx];
      __builtin_nontemporal_store(frac_act(x, xmin, scale), &out[idx]);
    }
  }
}

// ---------------------------------------------------------------------------
extern "C" void kernel_launch(void* const* d_in, const int* in_sizes, int n_in,
                              void* d_out, int out_size, void* d_ws, size_t ws_size,
                              hipStream_t stream) {
  const float* x = (const float*)d_in[0];
  float* out = (float*)d_out;
  unsigned* mm = (unsigned*)d_ws;  // mm[0]=min key, mm[1]=max key
  const long n = (long)in_sizes[0];

  fa_init_minmax<<<1, 32, 0, stream>>>(mm);

  // 2048 blocks * 256 thr = 16K waves; ~4K total same-address atomics,
  // spread across the kernel's lifetime.
  fa_minmax_reduce<<<2048, 256, 0, stream>>>(x, mm, n);

  const long tiles = (n >> 2) >> 8;
  long mb = tiles < 4096 ? (tiles > 0 ? tiles : 1) : 4096;
  fa_map<<<(int)mb, 256, 0, stream>>>(x, out, mm, n);
}